// GENConv_13039520711029
// MI455X (gfx1250) — compile-verified
//
#include <hip/hip_runtime.h>
#include <hip/hip_bf16.h>

typedef __attribute__((ext_vector_type(2))) float v2f;
typedef __attribute__((ext_vector_type(8))) float v8f;

#define D_IN   64
#define H_MID  128
#define D_OUT  64
#define GEPS   1e-7f
#define BNEPS  1e-5f
#define NRMEPS 1e-12f

// ---------------------------------------------------------------------------
// init: zero the accumulators (gmax, den, h, colsum, colsq)
// ---------------------------------------------------------------------------
__global__ void k_init(float* __restrict__ p, long n) {
    long i = (long)blockIdx.x * blockDim.x + threadIdx.x;
    long stride = (long)gridDim.x * blockDim.x;
    for (; i < n; i += stride) p[i] = 0.0f;
}

// helper: compute message float4 for edge e, channel block c4
__device__ __forceinline__ float4 msg4(const float* __restrict__ feats,
                                       const float* __restrict__ edge_h,
                                       int s, long e, int c4) {
    const float4 f  = *reinterpret_cast<const float4*>(feats  + (long)s * D_IN + c4);
    const float4 eh = *reinterpret_cast<const float4*>(edge_h + e * D_IN + c4);
    float4 m;
    m.x = fmaxf(f.x + eh.x, 0.0f) + GEPS;
    m.y = fmaxf(f.y + eh.y, 0.0f) + GEPS;
    m.z = fmaxf(f.z + eh.z, 0.0f) + GEPS;
    m.w = fmaxf(f.w + eh.w, 0.0f) + GEPS;
    return m;
}

// ---------------------------------------------------------------------------
// edge pass A: gmax[dst,c] = max over incoming edges of m  (m>0 -> int bits OK)
// ---------------------------------------------------------------------------
__global__ void k_edge_max(const float* __restrict__ feats,
                           const float* __restrict__ edge_h,
                           const int* __restrict__ src, const int* __restrict__ dst,
                           int* __restrict__ gmax_bits, long E) {
    long t = (long)blockIdx.x * blockDim.x + threadIdx.x;
    long e = t >> 4;
    if (e >= E) return;
    int c4 = (int)(t & 15) * 4;
    float4 m = msg4(feats, edge_h, src[e], e, c4);
    int* g = gmax_bits + (long)dst[e] * D_IN + c4;
    atomicMax(g + 0, __float_as_int(m.x));
    atomicMax(g + 1, __float_as_int(m.y));
    atomicMax(g + 2, __float_as_int(m.z));
    atomicMax(g + 3, __float_as_int(m.w));
}

// ---------------------------------------------------------------------------
// edge pass B: den[dst,c] += exp(m - gmax[dst,c])
// ---------------------------------------------------------------------------
__global__ void k_edge_den(const float* __restrict__ feats,
                           const float* __restrict__ edge_h,
                           const int* __restrict__ src, const int* __restrict__ dst,
                           const float* __restrict__ gmax,
                           float* __restrict__ den, long E) {
    long t = (long)blockIdx.x * blockDim.x + threadIdx.x;
    long e = t >> 4;
    if (e >= E) return;
    int c4 = (int)(t & 15) * 4;
    int d = dst[e];
    float4 m = msg4(feats, edge_h, src[e], e, c4);
    const float4 g = *reinterpret_cast<const float4*>(gmax + (long)d * D_IN + c4);
    float* dn = den + (long)d * D_IN + c4;
    atomicAdd(dn + 0, expf(m.x - g.x));
    atomicAdd(dn + 1, expf(m.y - g.y));
    atomicAdd(dn + 2, expf(m.z - g.z));
    atomicAdd(dn + 3, expf(m.w - g.w));
}

// ---------------------------------------------------------------------------
// edge pass C: h[dst,c] += m * exp(m-gmax)/den
// ---------------------------------------------------------------------------
__global__ void k_edge_agg(const float* __restrict__ feats,
                           const float* __restrict__ edge_h,
                           const int* __restrict__ src, const int* __restrict__ dst,
                           const float* __restrict__ gmax,
                           const float* __restrict__ den,
                           float* __restrict__ h, long E) {
    long t = (long)blockIdx.x * blockDim.x + threadIdx.x;
    long e = t >> 4;
    if (e >= E) return;
    int c4 = (int)(t & 15) * 4;
    int d = dst[e];
    float4 m = msg4(feats, edge_h, src[e], e, c4);
    const float4 g  = *reinterpret_cast<const float4*>(gmax + (long)d * D_IN + c4);
    const float4 dn = *reinterpret_cast<const float4*>(den  + (long)d * D_IN + c4);
    float* hp = h + (long)d * D_IN + c4;
    atomicAdd(hp + 0, m.x * (expf(m.x - g.x) / dn.x));
    atomicAdd(hp + 1, m.y * (expf(m.y - g.y) / dn.y));
    atomicAdd(hp + 2, m.z * (expf(m.z - g.z) / dn.z));
    atomicAdd(hp + 3, m.w * (expf(m.w - g.w) / dn.w));
}

// ---------------------------------------------------------------------------
// MessageNorm: x = feats + (h / max(||h||,eps)) * ||feats|| * scale
// one wave32 per node, 2 channels per lane
// ---------------------------------------------------------------------------
__global__ void k_msgnorm(const float* __restrict__ feats,
                          const float* __restrict__ h,
                          const float* __restrict__ scale,
                          float* __restrict__ x, int Nn) {
    int wave = threadIdx.x >> 5;
    int lane = threadIdx.x & 31;
    int node = blockIdx.x * 8 + wave;
    if (node >= Nn) return;
    long base = (long)node * D_IN;
    float f0 = feats[base + lane], f1 = feats[base + lane + 32];
    float h0 = h[base + lane],     h1 = h[base + lane + 32];
    float fs = f0 * f0 + f1 * f1;
    float hs = h0 * h0 + h1 * h1;
    #pragma unroll
    for (int off = 16; off > 0; off >>= 1) {
        fs += __shfl_xor(fs, off);
        hs += __shfl_xor(hs, off);
    }
    float fnorm = sqrtf(fs);
    float r = fnorm * scale[0] / fmaxf(sqrtf(hs), NRMEPS);
    x[base + lane]      = f0 + h0 * r;
    x[base + lane + 32] = f1 + h1 * r;
}

// ---------------------------------------------------------------------------
// GEMM1: y[N,128] = x[N,64] @ W1[64,128] + b1; fused per-column sum/sumsq.
// Block = 256 threads = 8 waves; block -> one 16-row strip; wave w -> cols 16w..
// WMMA f32 16x16x4:  A lane l: A[l%16][2*(l/16)+v], B lane l: B[2*(l/16)+v][l%16]
//                    C/D vgpr v: row v + 8*(l/16), col l%16
// ---------------------------------------------------------------------------
__global__ void k_gemm1(const float* __restrict__ x,
                        const float* __restrict__ W1,
                        const float* __restrict__ b1,
                        float* __restrict__ y,
                        float* __restrict__ colsum, float* __restrict__ colsq,
                        int Nn) {
    int wave = threadIdx.x >> 5;
    int lane = threadIdx.x & 31;
    int colBase = wave * 16;
    int r  = lane & 15;
    int kh = (lane >> 4) * 2;
    const float* arow = x + (long)(blockIdx.x * 16 + r) * D_IN;
    v8f c = {};
    #pragma unroll
    for (int kk = 0; kk < D_IN; kk += 4) {
        v2f a, b;
        a.x = arow[kk + kh];
        a.y = arow[kk + kh + 1];
        b.x = W1[(long)(kk + kh) * H_MID + colBase + r];
        b.y = W1[(long)(kk + kh + 1) * H_MID + colBase + r];
        c = __builtin_amdgcn_wmma_f32_16x16x4_f32(false, a, false, b,
                                                  (short)0, c, false, false);
    }
    float bias = b1[colBase + r];
    int rbase = blockIdx.x * 16 + (lane >> 4) * 8;
    float psum = 0.0f, psq = 0.0f;
    #pragma unroll
    for (int v = 0; v < 8; v++) {
        float val = c[v] + bias;
        y[(long)(rbase + v) * H_MID + colBase + r] = val;
        psum += val;
        psq  += val * val;
    }
    // combine the two half-waves (same column, rows 0-7 vs 8-15)
    psum += __shfl_xor(psum, 16);
    psq  += __shfl_xor(psq, 16);
    if (lane < 16) {
        atomicAdd(&colsum[colBase + r], psum);
        atomicAdd(&colsq[colBase + r],  psq);
    }
}

// ---------------------------------------------------------------------------
// fold BN batch stats into per-channel scale s and shift t
// ---------------------------------------------------------------------------
__global__ void k_bnfold(const float* __restrict__ colsum,
                         const float* __restrict__ colsq,
                         const float* __restrict__ gamma,
                         const float* __restrict__ bn_beta,
                         float* __restrict__ sArr, float* __restrict__ tArr,
                         int Nn) {
    int cix = threadIdx.x;
    if (cix >= H_MID) return;
    float invN = 1.0f / (float)Nn;
    float mean = colsum[cix] * invN;
    float var  = colsq[cix] * invN - mean * mean;
    float s = gamma[cix] * rsqrtf(var + BNEPS);
    sArr[cix] = s;
    tArr[cix] = bn_beta[cix] - mean * s;
}

// ---------------------------------------------------------------------------
// GEMM2: out[N,64] = relu(y*s + t) @ W2[128,64] + b2
// Block = 128 threads = 4 waves; block -> one 16-row strip; wave w -> cols 16w..
// ---------------------------------------------------------------------------
__global__ void k_gemm2(const float* __restrict__ y,
                        const float* __restrict__ W2,
                        const float* __restrict__ b2,
                        const float* __restrict__ sArr,
                        const float* __restrict__ tArr,
                        float* __restrict__ out) {
    int wave = threadIdx.x >> 5;
    int lane = threadIdx.x & 31;
    int colBase = wave * 16;
    int r  = lane & 15;
    int kh = (lane >> 4) * 2;
    const float* arow = y + (long)(blockIdx.x * 16 + r) * H_MID;
    v8f c = {};
    #pragma unroll
    for (int kk = 0; kk < H_MID; kk += 4) {
        int k0 = kk + kh;
        v2f a, b;
        a.x = fmaxf(arow[k0]     * sArr[k0]     + tArr[k0],     0.0f);
        a.y = fmaxf(arow[k0 + 1] * sArr[k0 + 1] + tArr[k0 + 1], 0.0f);
        b.x = W2[(long)k0 * D_OUT + colBase + r];
        b.y = W2[(long)(k0 + 1) * D_OUT + colBase + r];
        c = __builtin_amdgcn_wmma_f32_16x16x4_f32(false, a, false, b,
                                                  (short)0, c, false, false);
    }
    float bias = b2[colBase + r];
    int rbase = blockIdx.x * 16 + (lane >> 4) * 8;
    #pragma unroll
    for (int v = 0; v < 8; v++)
        out[(long)(rbase + v) * D_OUT + colBase + r] = c[v] + bias;
}

// ---------------------------------------------------------------------------
extern "C" void kernel_launch(void* const* d_in, const int* in_sizes, int n_in,
                              void* d_out, int out_size, void* d_ws, size_t ws_size,
                              hipStream_t stream) {
    const float* feats  = (const float*)d_in[0];
    const float* edge_h = (const float*)d_in[1];
    const int*   src    = (const int*)d_in[2];
    const int*   dst    = (const int*)d_in[3];
    const float* W1     = (const float*)d_in[4];
    const float* b1     = (const float*)d_in[5];
    const float* gamma  = (const float*)d_in[6];
    const float* bnbeta = (const float*)d_in[7];
    const float* W2     = (const float*)d_in[8];
    const float* b2     = (const float*)d_in[9];
    const float* scale  = (const float*)d_in[10];
    float* out = (float*)d_out;

    const int  Nn = in_sizes[0] / D_IN;   // 100000
    const long E  = in_sizes[1] / D_IN;   // 1600000

    // workspace layout (floats); gmax+den region is reused as y[N,128]
    float* gmax   = (float*)d_ws;                 // N*64
    float* den    = gmax + (long)Nn * D_IN;       // N*64
    float* hbuf   = den  + (long)Nn * D_IN;       // N*64
    float* xbuf   = hbuf + (long)Nn * D_IN;       // N*64
    float* ybuf   = gmax;                         // alias: N*128
    float* colsum = xbuf + (long)Nn * D_IN;       // 128
    float* colsq  = colsum + H_MID;               // 128
    float* sArr   = colsq + H_MID;                // 128
    float* tArr   = sArr + H_MID;                 // 128

    // zero accumulators: gmax, den, h (contiguous) + colsum, colsq
    long nz = (long)Nn * D_IN * 3;
    k_init<<<2048, 256, 0, stream>>>(gmax, nz);
    k_init<<<1, 256, 0, stream>>>(colsum, 2 * H_MID);

    long edgeThreads = E * 16;
    int  edgeBlocks  = (int)((edgeThreads + 255) / 256);
    k_edge_max<<<edgeBlocks, 256, 0, stream>>>(feats, edge_h, src, dst,
                                               (int*)gmax, E);
    k_edge_den<<<edgeBlocks, 256, 0, stream>>>(feats, edge_h, src, dst,
                                               gmax, den, E);
    k_edge_agg<<<edgeBlocks, 256, 0, stream>>>(feats, edge_h, src, dst,
                                               gmax, den, hbuf, E);

    k_msgnorm<<<(Nn + 7) / 8, 256, 0, stream>>>(feats, hbuf, scale, xbuf, Nn);

    int rowTiles = Nn / 16;  // 6250, exact
    k_gemm1<<<rowTiles, 256, 0, stream>>>(xbuf, W1, b1, ybuf, colsum, colsq, Nn);
    k_bnfold<<<1, H_MID, 0, stream>>>(colsum, colsq, gamma, bnbeta, sArr, tArr, Nn);
    k_gemm2<<<rowTiles, 128, 0, stream>>>(ybuf, W2, b2, sArr, tArr, out);
}